// ChebNet_257698037955
// MI455X (gfx1250) — compile-verified
//
#include <hip/hip_runtime.h>
#include <hip/hip_bf16.h>
#include <math.h>

typedef __attribute__((ext_vector_type(2))) float v2f;
typedef __attribute__((ext_vector_type(8))) float v8f;

// ---------------------------------------------------------------------------
// Setup / elementwise kernels
// ---------------------------------------------------------------------------

__global__ void k_fill_f32(float* __restrict__ p, int n, float v) {
  int i = blockIdx.x * blockDim.x + threadIdx.x;
  if (i < n) p[i] = v;
}

__global__ void k_fill_i32(int* __restrict__ p, int n, int v) {
  int i = blockIdx.x * blockDim.x + threadIdx.x;
  if (i < n) p[i] = v;
}

// deg[src] += w ; counts[dst] += 1   (one pass over edges)
__global__ void k_edge_stats(const int* __restrict__ ei, const float* __restrict__ ew,
                             float* __restrict__ deg, int* __restrict__ counts, int nE) {
  int e = blockIdx.x * blockDim.x + threadIdx.x;
  if (e < nE) {
    atomicAdd(&deg[ei[e]], ew[e]);
    atomicAdd(&counts[ei[nE + e]], 1);
  }
}

__global__ void k_dis(const float* __restrict__ deg, float* __restrict__ dis, int n) {
  int i = blockIdx.x * blockDim.x + threadIdx.x;
  if (i < n) {
    float d = deg[i];
    dis[i] = (d > 0.0f) ? rsqrtf(d) : 0.0f;
  }
}

// Single-block exclusive scan: row_start[0]=0, row_start[i+1]=sum(counts[0..i])
__global__ void k_scan(const int* __restrict__ counts, int* __restrict__ row_start, int n) {
  __shared__ int buf[256];
  __shared__ int carry;
  if (threadIdx.x == 0) { carry = 0; row_start[0] = 0; }
  __syncthreads();
  for (int base = 0; base < n; base += 256) {
    int i = base + (int)threadIdx.x;
    int v = (i < n) ? counts[i] : 0;
    buf[threadIdx.x] = v;
    __syncthreads();
    for (int off = 1; off < 256; off <<= 1) {
      int t = (threadIdx.x >= (unsigned)off) ? buf[threadIdx.x - off] : 0;
      __syncthreads();
      buf[threadIdx.x] += t;
      __syncthreads();
    }
    int inc = buf[threadIdx.x] + carry;
    if (i < n) row_start[i + 1] = inc;
    __syncthreads();
    if (threadIdx.x == 255) carry = inc;
    __syncthreads();
  }
}

// Scatter edges into CSR-by-dst; entry = (src, bitcast(norm)) packed in int2
// so the hot gather loop needs only one b64 load per neighbor.
__global__ void k_fill_csr(const int* __restrict__ ei, const float* __restrict__ ew,
                           const float* __restrict__ dis, const int* __restrict__ row_start,
                           int* __restrict__ cursor, int2* __restrict__ csr, int nE) {
  int e = blockIdx.x * blockDim.x + threadIdx.x;
  if (e < nE) {
    int s = ei[e], d = ei[nE + e];
    float nm = -dis[s] * ew[e] * dis[d];
    int pos = atomicAdd(&cursor[d], 1);
    csr[row_start[d] + pos] = make_int2(s, __float_as_int(nm));
  }
}

__global__ void k_silu(float* __restrict__ p, int n) {
  int i = blockIdx.x * blockDim.x + threadIdx.x;
  if (i < n) {
    float x = p[i];
    p[i] = x / (1.0f + expf(-x));
  }
}

// Final layer: K=1, [10000x30] @ [30x1], then sigmoid
__global__ void k_final(const float* __restrict__ h, const float* __restrict__ W4,
                        float* __restrict__ out, int n) {
  int i = blockIdx.x * blockDim.x + threadIdx.x;
  if (i < n) {
    float s = 0.0f;
#pragma unroll
    for (int j = 0; j < 30; ++j) s += h[i * 30 + j] * W4[j];
    out[i] = 1.0f / (1.0f + expf(-s));
  }
}

// ---------------------------------------------------------------------------
// Fused Chebyshev step:
//   do_prop=1:  T = scale * (L_hat @ gsrc)  [- tx0 if given]; tnext = T
//   do_prop=0:  T = gsrc                    (k == 0 case)
// then (WMMA, fp32):  out {=,+=} T_tile @ Wk [+ bias if init_out]
// Block = 16 nodes x 256 threads. Gather: thread(node=tid>>4, chunk=tid&15)
// accumulates float4 slices of neighbor rows (unrolled x2 for MLP).
// GEMM: waves 0..NT-1 each own a 16-col tile, CIN/4 chained
// V_WMMA_F32_16X16X4_F32 ops; epilogue is clause-friendly (loads, adds, stores)
// with tile-uniform guards only.
// ---------------------------------------------------------------------------
template <int CIN, int COUT>
__global__ __launch_bounds__(256) void cheb_step(
    const float* __restrict__ gsrc, const float* __restrict__ tx0,
    float* __restrict__ tnext,
    const int* __restrict__ row_start, const int2* __restrict__ csr,
    const float* __restrict__ Wk, const float* __restrict__ bias,
    float* __restrict__ out, int n, int do_prop, int init_out, float scale) {
  static_assert(CIN % 4 == 0 && CIN <= 64 && COUT <= 64, "shape limits");
  constexpr int NCH = CIN / 4;          // float4 chunks per row
  constexpr int NT = (COUT + 15) / 16;  // 16-wide output column tiles
  __shared__ __align__(16) float ldsA[16][CIN + 4];

  const int tid = threadIdx.x;
  const int nodeL = tid >> 4;  // 0..15 local node
  const int ch = tid & 15;     // float4 chunk id
  const int node0 = blockIdx.x * 16;
  const int node = node0 + nodeL;
  const bool rowOk = node < n;

  // ---- stage A tile (and T_k recurrence) ----
  if (ch < NCH) {
    float4 v = make_float4(0.0f, 0.0f, 0.0f, 0.0f);
    if (do_prop) {
      if (rowOk) {
        const int beg = row_start[node];
        const int end = row_start[node + 1];
        int j = beg;
        // unroll x2: two independent (b64 idx, b128 row) load pairs in flight
        for (; j + 1 < end; j += 2) {
          const int2 e0 = csr[j];
          const int2 e1 = csr[j + 1];
          const float w0 = __int_as_float(e0.y);
          const float w1 = __int_as_float(e1.y);
          const float4 h0 = *(const float4*)(gsrc + (size_t)e0.x * CIN + ch * 4);
          const float4 h1 = *(const float4*)(gsrc + (size_t)e1.x * CIN + ch * 4);
          v.x += w0 * h0.x + w1 * h1.x;
          v.y += w0 * h0.y + w1 * h1.y;
          v.z += w0 * h0.z + w1 * h1.z;
          v.w += w0 * h0.w + w1 * h1.w;
        }
        if (j < end) {
          const int2 e0 = csr[j];
          const float w0 = __int_as_float(e0.y);
          const float4 h0 = *(const float4*)(gsrc + (size_t)e0.x * CIN + ch * 4);
          v.x += w0 * h0.x;
          v.y += w0 * h0.y;
          v.z += w0 * h0.z;
          v.w += w0 * h0.w;
        }
        v.x *= scale; v.y *= scale; v.z *= scale; v.w *= scale;
        if (tx0) {  // T_k = 2 * L_hat T_{k-1} - T_{k-2}
          const float4 t0 = *(const float4*)(tx0 + (size_t)node * CIN + ch * 4);
          v.x -= t0.x; v.y -= t0.y; v.z -= t0.z; v.w -= t0.w;
        }
        *(float4*)(tnext + (size_t)node * CIN + ch * 4) = v;
      }
    } else {
      if (rowOk) v = *(const float4*)(gsrc + (size_t)node * CIN + ch * 4);
    }
    *(float4*)(&ldsA[nodeL][ch * 4]) = v;
  }
  __syncthreads();

  // ---- WMMA GEMM: out_tile {=,+=} A(16xCIN) @ Wk(CINxCOUT) ----
  const int wid = tid >> 5;  // wave32
  if (wid < NT) {            // wave-uniform branch: EXEC all ones for WMMA
    const int lane = tid & 31;
    const int lh = lane >> 4;  // lane half
    const int lm = lane & 15;
    const int col = wid * 16 + lm;
    const bool colOk = col < COUT;

    v8f c;
    float cinit = 0.0f;
    if (init_out && bias != nullptr && colOk) cinit = bias[col];
#pragma unroll
    for (int r = 0; r < 8; ++r) c[r] = cinit;

#pragma unroll
    for (int ks = 0; ks < CIN / 4; ++ks) {
      const int k0 = ks * 4 + lh * 2;
      v2f a, b;
      // A 16x4 f32 layout: lanes 0-15 -> M=lane, VGPR0/1 = K0/K1; lanes 16-31 -> K2/K3
      a[0] = ldsA[lm][k0];
      a[1] = ldsA[lm][k0 + 1];
      // B 4x16 f32 layout (mirror of A): lane half selects K pair, lane%16 = N
      b[0] = colOk ? Wk[(size_t)(k0) * COUT + col] : 0.0f;
      b[1] = colOk ? Wk[(size_t)(k0 + 1) * COUT + col] : 0.0f;
      c = __builtin_amdgcn_wmma_f32_16x16x4_f32(false, a, false, b, (short)0, c,
                                                false, false);
    }

    // C/D 16x16 f32 layout: lane<16 -> N=lane, VGPR r = M=r; lane>=16 -> M=8+r
    const bool fullTile = (node0 + 16 <= n);
    if (colOk) {
      float* base = out + (size_t)(node0 + lh * 8) * COUT + col;
      if (init_out) {
        if (fullTile) {
#pragma unroll
          for (int r = 0; r < 8; ++r) base[(size_t)r * COUT] = c[r];
        } else {
#pragma unroll
          for (int r = 0; r < 8; ++r)
            if (node0 + lh * 8 + r < n) base[(size_t)r * COUT] = c[r];
        }
      } else {
        if (fullTile) {
          float oldv[8];
#pragma unroll
          for (int r = 0; r < 8; ++r) oldv[r] = base[(size_t)r * COUT];
#pragma unroll
          for (int r = 0; r < 8; ++r) base[(size_t)r * COUT] = oldv[r] + c[r];
        } else {
#pragma unroll
          for (int r = 0; r < 8; ++r)
            if (node0 + lh * 8 + r < n) base[(size_t)r * COUT] += c[r];
        }
      }
    }
  }
}

// ---------------------------------------------------------------------------
// Host-side driver for one ChebConv layer
// ---------------------------------------------------------------------------
template <int CIN, int COUT>
static void run_layer(const float* h, const float* W, const float* b, int K,
                      float* out, float* Ta, float* Tb, float* Tc,
                      const int* row_start, const int2* csr,
                      int n, hipStream_t stream) {
  dim3 grid((n + 15) / 16), blk(256);
  // k = 0: out = h @ W0 + b
  cheb_step<CIN, COUT><<<grid, blk, 0, stream>>>(h, nullptr, nullptr, row_start,
      csr, W, b, out, n, /*do_prop=*/0, /*init_out=*/1, 1.0f);
  if (K <= 1) return;
  // k = 1: T1 = L_hat h ; out += T1 @ W1
  cheb_step<CIN, COUT><<<grid, blk, 0, stream>>>(h, nullptr, Ta, row_start,
      csr, W + (size_t)CIN * COUT, nullptr, out, n, 1, 0, 1.0f);
  float* bufs[3] = {Ta, Tb, Tc};
  int i0 = -1, i1 = 0;  // tx0 = h (i0 < 0), tx1 = Ta
  for (int k = 2; k < K; ++k) {
    int in_ = (i0 < 0) ? ((i1 + 1) % 3) : (3 - i0 - i1);
    const float* tx0 = (i0 < 0) ? h : bufs[i0];
    cheb_step<CIN, COUT><<<grid, blk, 0, stream>>>(bufs[i1], tx0, bufs[in_],
        row_start, csr, W + (size_t)k * CIN * COUT, nullptr, out,
        n, 1, 0, 2.0f);
    i0 = i1;
    i1 = in_;
  }
}

// ---------------------------------------------------------------------------
// Entry point
// ---------------------------------------------------------------------------
extern "C" void kernel_launch(void* const* d_in, const int* in_sizes, int n_in,
                              void* d_out, int out_size, void* d_ws, size_t ws_size,
                              hipStream_t stream) {
  const float* x  = (const float*)d_in[0];   // [n, 4]
  const int*   ei = (const int*)d_in[1];     // [2, nE]
  const float* ew = (const float*)d_in[2];   // [nE]
  const float* W1 = (const float*)d_in[3];   // [120, 4, 64]
  const float* b1 = (const float*)d_in[4];   // [64]
  const float* W2 = (const float*)d_in[5];   // [120, 64, 60]
  const float* b2 = (const float*)d_in[6];   // [60]
  const float* W3 = (const float*)d_in[7];   // [20, 60, 30]
  const float* b3 = (const float*)d_in[8];   // [30]
  const float* W4 = (const float*)d_in[9];   // [1, 30, 1]
  float* out = (float*)d_out;

  const int n  = in_sizes[0] / 4;
  const int nE = in_sizes[2];

  // Workspace carve (~16 MB total for n=10000, nE=320000)
  uintptr_t p = (uintptr_t)d_ws;
  auto carve = [&](size_t bytes) -> void* {
    void* r = (void*)p;
    p += (bytes + 255) & ~((size_t)255);
    return r;
  };
  float* deg       = (float*)carve(sizeof(float) * (size_t)n);
  float* dis       = (float*)carve(sizeof(float) * (size_t)n);
  int*   counts    = (int*)carve(sizeof(int) * (size_t)n);
  int*   cursor    = (int*)carve(sizeof(int) * (size_t)n);
  int*   row_start = (int*)carve(sizeof(int) * (size_t)(n + 1));
  int2*  csr       = (int2*)carve(sizeof(int2) * (size_t)nE);
  float* Ta        = (float*)carve(sizeof(float) * (size_t)n * 64);
  float* Tb        = (float*)carve(sizeof(float) * (size_t)n * 64);
  float* Tc        = (float*)carve(sizeof(float) * (size_t)n * 64);
  float* hA        = (float*)carve(sizeof(float) * (size_t)n * 64);
  float* hB        = (float*)carve(sizeof(float) * (size_t)n * 64);
  (void)n_in; (void)out_size; (void)ws_size;

  const int nb = (n + 255) / 256;
  const int eb = (nE + 255) / 256;

  // Graph preprocessing (every call; deterministic structure, L2-resident)
  k_fill_f32<<<nb, 256, 0, stream>>>(deg, n, 0.0f);
  k_fill_i32<<<nb, 256, 0, stream>>>(counts, n, 0);
  k_fill_i32<<<nb, 256, 0, stream>>>(cursor, n, 0);
  k_edge_stats<<<eb, 256, 0, stream>>>(ei, ew, deg, counts, nE);
  k_dis<<<nb, 256, 0, stream>>>(deg, dis, n);
  k_scan<<<1, 256, 0, stream>>>(counts, row_start, n);
  k_fill_csr<<<eb, 256, 0, stream>>>(ei, ew, dis, row_start, cursor, csr, nE);

  // Layer 1: ChebConv(4 -> 64, K=120) + SiLU
  run_layer<4, 64>(x, W1, b1, 120, hA, Ta, Tb, Tc, row_start, csr, n, stream);
  k_silu<<<(n * 64 + 255) / 256, 256, 0, stream>>>(hA, n * 64);
  // Layer 2: ChebConv(64 -> 60, K=120) + SiLU
  run_layer<64, 60>(hA, W2, b2, 120, hB, Ta, Tb, Tc, row_start, csr, n, stream);
  k_silu<<<(n * 60 + 255) / 256, 256, 0, stream>>>(hB, n * 60);
  // Layer 3: ChebConv(60 -> 30, K=20) + SiLU
  run_layer<60, 30>(hB, W3, b3, 20, hA, Ta, Tb, Tc, row_start, csr, n, stream);
  k_silu<<<(n * 30 + 255) / 256, 256, 0, stream>>>(hA, n * 30);
  // Layer 4: ChebConv(30 -> 1, K=1) + sigmoid
  k_final<<<nb, 256, 0, stream>>>(hA, W4, out, n);
}